// QIntSoftmaxShift_21234318311677
// MI455X (gfx1250) — compile-verified
//
#include <hip/hip_runtime.h>
#include <math.h>

// ---------------------------------------------------------------------------
// shiftmax (integer-only softmax approx) over a 29.8M-element f32 tensor.
// Memory-bound: ~119MB tensor, fits in MI455X 192MB L2 -> pass1 reads HBM,
// passes 2/3 read L2, final output written with non-temporal b128 stores.
// Global reductions done in two deterministic stages through d_ws.
// Wave-level sum reduction uses v_wmma_f32_16x16x4_f32 (ones-matrix trick).
// exp2 uses raw v_exp_f32 (exponent 20-q never reaches subnormal range for
// data that contributes non-negligibly), trimming ~5 VALU/element vs exp2f.
// ---------------------------------------------------------------------------

#define TPB 256
#define NB  1024
#define NWAVES (TPB / 32)

typedef __attribute__((ext_vector_type(2))) float v2f;
typedef __attribute__((ext_vector_type(4))) float v4f;
typedef __attribute__((ext_vector_type(8))) float v8f;

// raw v_exp_f32 (no subnormal-range fixup)
__device__ __forceinline__ float fast_exp2(float x) {
#if __has_builtin(__builtin_amdgcn_exp2f)
    return __builtin_amdgcn_exp2f(x);
#else
    return exp2f(x);
#endif
}

// ---------------- wave-level reductions (wave32) ----------------

__device__ __forceinline__ float wave_max32(float v) {
#pragma unroll
    for (int off = 16; off >= 1; off >>= 1)
        v = fmaxf(v, __shfl_xor(v, off, 32));
    return v;
}

__device__ __forceinline__ float wave_sum32(float v) {
#if __has_builtin(__builtin_amdgcn_wmma_f32_16x16x4_f32)
    // A (16x4): lane m vgpr0 = A[m][0] = v_m ; lane m+16 vgpr0 = A[m][2] = v_{m+16};
    // vgpr1 (= a[1]) zero. B = ones(4x16)  =>  D[m][n] = v_m + v_{m+16}.
    // D row layout: lane n, vgpr j holds M=j (lanes 0-15) / M=j+8 (lanes 16-31),
    // so summing the 8 accumulator regs gives the half-sum; one xor-16 finishes.
    v2f a; a[0] = v;    a[1] = 0.0f;
    v2f b; b[0] = 1.0f; b[1] = 1.0f;
    v8f c = {};
    v8f d = __builtin_amdgcn_wmma_f32_16x16x4_f32(false, a, false, b,
                                                  (short)0, c, false, false);
    float s = 0.0f;
#pragma unroll
    for (int i = 0; i < 8; ++i) s += d[i];
    s += __shfl_xor(s, 16, 32);
    return s;
#else
#pragma unroll
    for (int off = 16; off >= 1; off >>= 1)
        v += __shfl_xor(v, off, 32);
    return v;
#endif
}

// block reductions: result valid in threadIdx.x == 0
__device__ __forceinline__ float block_max(float v, float* smem) {
    const int lane = threadIdx.x & 31, wid = threadIdx.x >> 5;
    v = wave_max32(v);
    if (lane == 0) smem[wid] = v;
    __syncthreads();
    float r = -INFINITY;
    if (threadIdx.x == 0) {
        r = smem[0];
#pragma unroll
        for (int i = 1; i < NWAVES; ++i) r = fmaxf(r, smem[i]);
    }
    return r;
}

__device__ __forceinline__ float block_sum(float v, float* smem) {
    const int lane = threadIdx.x & 31, wid = threadIdx.x >> 5;
    v = wave_sum32(v);   // all 32 lanes active: EXEC==all-ones for WMMA
    if (lane == 0) smem[wid] = v;
    __syncthreads();
    float r = 0.0f;
    if (threadIdx.x == 0) {
#pragma unroll
        for (int i = 0; i < NWAVES; ++i) r += smem[i];
    }
    return r;
}

// ---------------- element math ----------------

__device__ __forceinline__ float iexp_elem(float xi, float inv_s, float mx,
                                           float I0, float qinv) {
    float Ip = (xi * inv_s - mx) * 1.4375f;   // I_delta * (1 + 0.5 - 0.0625)
    float q  = floorf(Ip * qinv);             // floor(Ip / (-I0))
    float r  = -(Ip + q * I0);
    float Ib = -r * 0.5f + I0;
    return Ib * fast_exp2(20.0f - q);         // Ib * 2^(N - q), N = 20
}

// ---------------- kernels ----------------

__global__ __launch_bounds__(TPB)
void kmax_partial(const float* __restrict__ x, const float* __restrict__ sc,
                  float* __restrict__ pmax, long long n) {
    __shared__ float smem[NWAVES];
    const float inv_s = 1.0f / sc[0];
    const long long nvec   = n >> 2;
    const long long stride = (long long)gridDim.x * blockDim.x;
    const long long gtid   = (long long)blockIdx.x * blockDim.x + threadIdx.x;

    float m = -INFINITY;
    for (long long i = gtid; i < nvec; i += stride) {
        v4f v = ((const v4f*)x)[i];                    // global_load_b128
        m = fmaxf(m, fmaxf(fmaxf(v.x * inv_s, v.y * inv_s),
                           fmaxf(v.z * inv_s, v.w * inv_s)));
    }
    long long t = (nvec << 2) + gtid;                  // scalar tail (n%4)
    if (t < n) m = fmaxf(m, x[t] * inv_s);

    float r = block_max(m, smem);
    if (threadIdx.x == 0) pmax[blockIdx.x] = r;
}

__global__ __launch_bounds__(TPB)
void kmax_final(const float* __restrict__ pmax, float* __restrict__ gmax, int nb) {
    __shared__ float smem[NWAVES];
    float m = -INFINITY;
    for (int i = threadIdx.x; i < nb; i += blockDim.x) m = fmaxf(m, pmax[i]);
    float r = block_max(m, smem);
    if (threadIdx.x == 0) gmax[0] = r;
}

__global__ __launch_bounds__(TPB)
void ksum_partial(const float* __restrict__ x, const float* __restrict__ sc,
                  const float* __restrict__ gmax, float* __restrict__ psum,
                  long long n) {
    __shared__ float smem[NWAVES];
    const float s0    = sc[0];
    const float inv_s = 1.0f / s0;
    const float I0    = rintf(1.0f / s0);      // round-half-even == jnp.round
    const float qinv  = 1.0f / (-I0);
    const float mx    = gmax[0];

    const long long nvec   = n >> 2;
    const long long stride = (long long)gridDim.x * blockDim.x;
    const long long gtid   = (long long)blockIdx.x * blockDim.x + threadIdx.x;

    float acc = 0.0f;
    for (long long i = gtid; i < nvec; i += stride) {
        v4f v = ((const v4f*)x)[i];                    // L2-resident re-read
        acc += iexp_elem(v.x, inv_s, mx, I0, qinv);
        acc += iexp_elem(v.y, inv_s, mx, I0, qinv);
        acc += iexp_elem(v.z, inv_s, mx, I0, qinv);
        acc += iexp_elem(v.w, inv_s, mx, I0, qinv);
    }
    long long t = (nvec << 2) + gtid;
    if (t < n) acc += iexp_elem(x[t], inv_s, mx, I0, qinv);

    float r = block_sum(acc, smem);                    // v_wmma reduction
    if (threadIdx.x == 0) psum[blockIdx.x] = r;
}

__global__ __launch_bounds__(TPB)
void ksum_final(const float* __restrict__ psum, float* __restrict__ factor, int nb) {
    __shared__ float smem[NWAVES];
    float acc = 0.0f;
    for (int i = threadIdx.x; i < nb; i += blockDim.x) acc += psum[i];
    float r = block_sum(acc, smem);
    if (threadIdx.x == 0) {
        // floor(2^30/denom) * 2^(-30+7) * 2^-7 == floor(2^30/denom) * 2^-30
        factor[0] = floorf(1073741824.0f / r) * 0x1p-30f;
    }
}

__global__ __launch_bounds__(TPB)
void kout(const float* __restrict__ x, const float* __restrict__ sc,
          const float* __restrict__ gmax, const float* __restrict__ factor,
          float* __restrict__ out, long long n) {
    const float s0    = sc[0];
    const float inv_s = 1.0f / s0;
    const float I0    = rintf(1.0f / s0);
    const float qinv  = 1.0f / (-I0);
    const float mx    = gmax[0];
    const float fac   = factor[0];

    const long long nvec   = n >> 2;
    const long long stride = (long long)gridDim.x * blockDim.x;
    const long long gtid   = (long long)blockIdx.x * blockDim.x + threadIdx.x;

    for (long long i = gtid; i < nvec; i += stride) {
        v4f v = ((const v4f*)x)[i];
        v4f o;
        o.x = fac * iexp_elem(v.x, inv_s, mx, I0, qinv);
        o.y = fac * iexp_elem(v.y, inv_s, mx, I0, qinv);
        o.z = fac * iexp_elem(v.z, inv_s, mx, I0, qinv);
        o.w = fac * iexp_elem(v.w, inv_s, mx, I0, qinv);
        __builtin_nontemporal_store(o, ((v4f*)out) + i);   // NT b128 store
    }
    long long t = (nvec << 2) + gtid;
    if (t < n) {
        float o = fac * iexp_elem(x[t], inv_s, mx, I0, qinv);
        __builtin_nontemporal_store(o, out + t);
    }
}

// ---------------- launch ----------------

extern "C" void kernel_launch(void* const* d_in, const int* in_sizes, int n_in,
                              void* d_out, int out_size, void* d_ws, size_t ws_size,
                              hipStream_t stream) {
    const float* x  = (const float*)d_in[0];
    const float* sc = (const float*)d_in[1];
    float* out = (float*)d_out;
    const long long n = (long long)in_sizes[0];

    float* w      = (float*)d_ws;
    float* gmax   = w;            // 1 float
    float* factor = w + 1;        // 1 float
    float* pmax   = w + 2;        // NB floats
    float* psum   = w + 2 + NB;   // NB floats

    kmax_partial<<<NB, TPB, 0, stream>>>(x, sc, pmax, n);
    kmax_final  <<<1,  TPB, 0, stream>>>(pmax, gmax, NB);
    ksum_partial<<<NB, TPB, 0, stream>>>(x, sc, gmax, psum, n);
    ksum_final  <<<1,  TPB, 0, stream>>>(psum, factor, NB);
    kout        <<<NB, TPB, 0, stream>>>(x, sc, gmax, factor, out, n);
}